// _ScaledDotProductAttention_2164663517287
// MI455X (gfx1250) — compile-verified
//
#include <hip/hip_runtime.h>

typedef __attribute__((ext_vector_type(2))) float v2f;
typedef __attribute__((ext_vector_type(8))) float v8f;

namespace {
constexpr int Bc = 4, Hc = 8, Lc = 2048, Dc = 64;
constexpr int BM  = 32;          // rows per workgroup
constexpr int BN  = 64;          // columns per tile iteration
constexpr int NCI = Lc / BN;     // 32 column iterations
constexpr int NRB = Lc / BM;     // 64 row-blocks per (b,h)
constexpr float SCALE = 0.125f;  // D^-0.5
constexpr float LOG2E = 1.4426950408889634f;
constexpr int SSTR = 2052;       // stash row stride (dwords), padded for banks
constexpr int KSTR = 80;         // K/V tile row stride (dwords), padded for banks
}

// CDNA5 async Global->LDS copy (ASYNCcnt-tracked), per ISA 15.18.3 op 98.
__device__ __forceinline__ void async_ld_b128(const float* lds_dst, unsigned goff_bytes,
                                              const void* gbase) {
    unsigned lds_addr = (unsigned)(size_t)lds_dst;   // low 32b of shared-aperture ptr = LDS offset
    asm volatile("global_load_async_to_lds_b128 %0, %1, %2"
                 :: "v"(lds_addr), "v"(goff_bytes), "s"(gbase)
                 : "memory");
}
__device__ __forceinline__ void wait_async0() {
    asm volatile("s_wait_asynccnt 0x0" ::: "memory");
}

__global__ __launch_bounds__(256)
void attn_fused_kernel(const float* __restrict__ qg, const float* __restrict__ kg,
                       const float* __restrict__ vg, const float* __restrict__ pg,
                       float* __restrict__ og, float* __restrict__ wgt,
                       float* __restrict__ sg)
{
    __shared__ float kvbuf[2][64 * KSTR];   // 2 x 20 KB double-buffered K/V tiles
    __shared__ float rowmax[BM];
    __shared__ float rowinv[BM];
    extern __shared__ float stash[];        // BM * SSTR floats (262 KB dynamic LDS)

    const int tid  = threadIdx.x;
    const int lane = tid & 31;
    const int wv   = tid >> 5;              // wave 0..7
    const int tr   = wv >> 2;               // tile row 0..1
    const int tc   = wv & 3;                // tile col 0..3
    const int bh   = blockIdx.x / NRB;
    const int rb   = blockIdx.x % NRB;
    const int r0   = rb * BM;

    const float* qb = qg + (size_t)bh * Lc * Dc;
    const float* kb = kg + (size_t)bh * Dc * Lc;   // k is [D, L]
    const float* vb = vg + (size_t)bh * Lc * Dc;
    const float* pb = pg + (size_t)bh * Lc * Lc;
    float* ob = og  + (size_t)bh * Lc * Dc;
    float* wb = wgt + (size_t)bh * Lc * Lc;
    float* sb = sg  + (size_t)bh * Lc * Lc;

    const int hs = (lane < 16) ? 0 : 2;     // K-half select per 16x16x4 operand layout
    const int ln = lane & 15;

    // Per-thread staging slot: 4 b128 transfers cover one 64x64 tile.
    const int s_rr = tid >> 4;              // base row 0..15 (stride 16 below)
    const int s_cg = (tid & 15) << 2;       // col group

    // ---- preload A (Q) operands for this wave: 16 rows x 64 K in 16 v2f regs ----
    v2f aq[16];
    {
        const float* qr = qb + (size_t)(r0 + tr*16 + ln) * Dc + hs;
        #pragma unroll
        for (int kk = 0; kk < 16; ++kk) {
            aq[kk].x = qr[4*kk + 0];
            aq[kk].y = qr[4*kk + 1];
        }
    }

    // ---- phase 1: S = Q*K * scale + prev ; write scores, stash in LDS ----
    auto stageK = [&](int ci, int b) {
        const int c0 = ci * BN;
        #pragma unroll
        for (int i = 0; i < 4; ++i) {
            const int rr = s_rr + 16*i;
            async_ld_b128(&kvbuf[b][rr*KSTR + s_cg],
                          (unsigned)((rr*Lc + c0 + s_cg) * 4), kb);
        }
    };

    stageK(0, 0);
    wait_async0();
    __syncthreads();

    for (int ci = 0; ci < NCI; ++ci) {
        const int cur = ci & 1;
        if (ci + 1 < NCI) {
            stageK(ci + 1, 1 - cur);                    // overlap with compute
            __builtin_prefetch(&pb[(size_t)(r0 + lane) * Lc + ci*BN + BN], 0, 0);
        }

        // Batch-load all B operands, then run the WMMA chain back-to-back.
        v2f bq[16];
        #pragma unroll
        for (int kk = 0; kk < 16; ++kk) {
            const int kr = 4*kk + hs;
            bq[kk].x = kvbuf[cur][kr*KSTR     + tc*16 + ln];
            bq[kk].y = kvbuf[cur][(kr+1)*KSTR + tc*16 + ln];
        }
        v8f acc = {};
        #pragma unroll
        for (int kk = 0; kk < 16; ++kk)
            acc = __builtin_amdgcn_wmma_f32_16x16x4_f32(false, aq[kk], false, bq[kk],
                                                        (short)0, acc, false, false);

        const int nc = ci*BN + tc*16 + ln;
        #pragma unroll
        for (int i = 0; i < 8; ++i) {
            const int mr = tr*16 + i + ((lane < 16) ? 0 : 8);
            const float s = acc[i]*SCALE + pb[(size_t)(r0+mr)*Lc + nc];
            sb[(size_t)(r0+mr)*Lc + nc] = s;
            stash[mr*SSTR + nc] = s;
        }

        wait_async0();                                  // next tile landed in LDS
        __syncthreads();
    }

    // ---- phase 1.5: softmax stats; stash := exp(s - rowmax) (un-normalized) ----
    {
        const int row  = tid >> 3;                      // 8 threads per row
        const int part = tid & 7;
        float4* s4 = (float4*)&stash[row*SSTR + part*256];
        float m = -1e30f;
        for (int c = 0; c < 64; ++c) {
            const float4 sv = s4[c];
            m = fmaxf(m, fmaxf(fmaxf(sv.x, sv.y), fmaxf(sv.z, sv.w)));
        }
        m = fmaxf(m, __shfl_xor(m, 1, 32));
        m = fmaxf(m, __shfl_xor(m, 2, 32));
        m = fmaxf(m, __shfl_xor(m, 4, 32));
        if (part == 0) rowmax[row] = m;
        __syncthreads();

        const float rm = rowmax[row];
        float sum = 0.f;
        for (int c = 0; c < 64; ++c) {                  // exclusive slice: safe in-place
            float4 sv = s4[c];
            sv.x = exp2f((sv.x - rm)*LOG2E);
            sv.y = exp2f((sv.y - rm)*LOG2E);
            sv.z = exp2f((sv.z - rm)*LOG2E);
            sv.w = exp2f((sv.w - rm)*LOG2E);
            s4[c] = sv;
            sum += (sv.x + sv.y) + (sv.z + sv.w);
        }
        sum += __shfl_xor(sum, 1, 32);
        sum += __shfl_xor(sum, 2, 32);
        sum += __shfl_xor(sum, 4, 32);
        if (part == 0) rowinv[row] = 1.0f / sum;
        __syncthreads();
    }

    // ---- phase 2a: weights = stash * inv_sum, in-place + HBM write ----
    for (int it = 0; it < 64; ++it) {
        const int idx4 = tid + 256*it;                  // 16384 float4s total
        const int e    = idx4 << 2;
        const int row  = e >> 11;
        const int col  = e & 2047;
        const float4 sv = *(const float4*)&stash[row*SSTR + col];
        const float ri = rowinv[row];
        float4 w4;
        w4.x = sv.x * ri;
        w4.y = sv.y * ri;
        w4.z = sv.z * ri;
        w4.w = sv.w * ri;
        *(float4*)&stash[row*SSTR + col] = w4;
        *(float4*)(wb + (size_t)(r0+row)*Lc + col) = w4;
    }
    __syncthreads();

    // ---- phase 2b: O = W * V (A from weight stash, B from async-staged V tiles) ----
    auto stageV = [&](int ci, int b) {
        #pragma unroll
        for (int i = 0; i < 4; ++i) {
            const int rr = s_rr + 16*i;
            async_ld_b128(&kvbuf[b][rr*KSTR + s_cg],
                          (unsigned)(((ci*64 + rr)*Dc + s_cg) * 4), vb);
        }
    };

    stageV(0, 0);
    wait_async0();
    __syncthreads();

    v8f oacc = {};
    for (int ci = 0; ci < NCI; ++ci) {
        const int cur = ci & 1;
        if (ci + 1 < NCI) stageV(ci + 1, 1 - cur);      // overlap with compute

        v2f aw[16], bv[16];
        #pragma unroll
        for (int kk = 0; kk < 16; ++kk) {
            const int kc = ci*64 + 4*kk + hs;
            aw[kk].x = stash[(tr*16 + ln)*SSTR + kc];
            aw[kk].y = stash[(tr*16 + ln)*SSTR + kc + 1];
            const int kr = 4*kk + hs;
            bv[kk].x = kvbuf[cur][kr*KSTR     + tc*16 + ln];
            bv[kk].y = kvbuf[cur][(kr+1)*KSTR + tc*16 + ln];
        }
        #pragma unroll
        for (int kk = 0; kk < 16; ++kk)
            oacc = __builtin_amdgcn_wmma_f32_16x16x4_f32(false, aw[kk], false, bv[kk],
                                                         (short)0, oacc, false, false);

        wait_async0();
        __syncthreads();
    }
    #pragma unroll
    for (int i = 0; i < 8; ++i) {
        const int mr = tr*16 + i + ((lane < 16) ? 0 : 8);
        ob[(size_t)(r0+mr)*Dc + tc*16 + ln] = oacc[i];
    }
}

extern "C" void kernel_launch(void* const* d_in, const int* in_sizes, int n_in,
                              void* d_out, int out_size, void* d_ws, size_t ws_size,
                              hipStream_t stream) {
    (void)in_sizes; (void)n_in; (void)out_size; (void)d_ws; (void)ws_size;
    const float* q    = (const float*)d_in[0];
    const float* k    = (const float*)d_in[1];
    const float* v    = (const float*)d_in[2];
    const float* prev = (const float*)d_in[3];

    float* out  = (float*)d_out;
    float* wout = out  + (size_t)Bc * Hc * Lc * Dc;   // attn_weights after output
    float* sout = wout + (size_t)Bc * Hc * Lc * Lc;   // attn_scores last

    const size_t dynLds = (size_t)BM * SSTR * sizeof(float);  // 262,656 B
    dim3 grid(Bc * Hc * NRB);   // 2048 workgroups, same-(b,h) blocks adjacent for L2 reuse
    attn_fused_kernel<<<grid, 256, dynLds, stream>>>(q, k, v, prev, out, wout, sout);
}